// StrucTreeEncoder_3770981286657
// MI455X (gfx1250) — compile-verified
//
#include <hip/hip_runtime.h>
#include <hip/hip_bf16.h>
#include <stdint.h>

typedef float        v2f __attribute__((ext_vector_type(2)));
typedef float        v8f __attribute__((ext_vector_type(8)));
typedef unsigned int v4u __attribute__((ext_vector_type(4)));
typedef int          v4i __attribute__((ext_vector_type(4)));
typedef int          v8i __attribute__((ext_vector_type(8)));

#define NNODE 524288
#define BSZ   256
#define NB    (NNODE / BSZ)

// LDS layout (in floats): one flat array so every access is a single ds op.
#define X_OFF      0      // 2 buffers x 512 floats   (x tiles, 256 nodes x 2)
#define HD_OFF     1024   // 2 buffers x 4096 floats  (hd tiles, 256 nodes x 16)
#define DUMMY_OFF  9216   // 32-float scatter sink for non-canonical lanes
#define LDS_FLOATS 9248

// ---------------- helpers ----------------

__device__ __forceinline__ float sigm(float v) {
  // sigmoid(v) = 1 / (1 + 2^(-v*log2(e)))  -> v_exp_f32 + v_rcp_f32
  float e = __builtin_amdgcn_exp2f(-1.44269504088896f * v);
  return __builtin_amdgcn_rcpf(1.0f + e);
}

// swap half-waves (lanes 0-15 <-> 16-31): ds_swizzle xor=0x10, and=0x1f
__device__ __forceinline__ float swap16(float v) {
  return __int_as_float(__builtin_amdgcn_ds_swizzle(__float_as_int(v), 0x401F));
}

// 16-vec (B-layout, 8 regs) x 16x16 weight (A-layout, 4 chunks) + C(D-layout)
__device__ __forceinline__ v8f wmma4(const v2f* A, const float* B, v8f c) {
#pragma unroll
  for (int k = 0; k < 4; ++k) {
    v2f b; b.x = B[2 * k]; b.y = B[2 * k + 1];
    c = __builtin_amdgcn_wmma_f32_16x16x4_f32(false, A[k], false, b,
                                              (short)0, c, false, false);
  }
  return c;
}

// sigmoid of D-layout acc -> sD (D-layout) and B (B-layout for next matvec)
__device__ __forceinline__ void sigB(const v8f& acc, bool lo, float* sD, float* B) {
  float sw[8];
#pragma unroll
  for (int r = 0; r < 8; ++r) sD[r] = sigm(acc[r]);
#pragma unroll
  for (int r = 0; r < 8; ++r) sw[r] = swap16(sD[r]);
  // b[2c+p]: lanes0-15 = v[4c+p], lanes16-31 = v[4c+p+2]
  B[0] = lo ? sD[0] : sw[2];
  B[1] = lo ? sD[1] : sw[3];
  B[2] = lo ? sD[4] : sw[6];
  B[3] = lo ? sD[5] : sw[7];
  B[4] = lo ? sw[0] : sD[2];
  B[5] = lo ? sw[1] : sD[3];
  B[6] = lo ? sw[4] : sD[6];
  B[7] = lo ? sw[5] : sD[7];
}

// ---- Tensor Data Mover: 1-D copy of ndw dwords (as a 2D tile ndw x 1) ----
__device__ __forceinline__ void tdm_desc(uint32_t lds_byte, uint64_t ga,
                                         uint32_t ndw, v4u& g0, v8i& g1,
                                         v4i& g2, v4i& g3) {
  g0[0] = 1u;                                   // count=1 (valid descriptor)
  g0[1] = lds_byte;                             // lds_addr
  g0[2] = (uint32_t)ga;                         // global_addr[31:0]
  g0[3] = (uint32_t)(ga >> 32) | 0x80000000u;   // global_addr[56:32] | type=2
  g1[0] = (int)0x00020000u;                     // data_size=2 (4 bytes)
  g1[1] = (int)((ndw & 0xFFFFu) << 16);         // tensor_dim0 lo16
  g1[2] = (int)(((ndw >> 16) & 0xFFFFu) | (1u << 16)); // dim0 hi16 | tensor_dim1=1
  g1[3] = (int)((ndw & 0xFFFFu) << 16);         // tensor_dim1 hi | tile_dim0=ndw
  g1[4] = 1;                                    // tile_dim1=1, tile_dim2=0
  g1[5] = (int)ndw;                             // tensor_dim0_stride lo32
  g1[6] = 0;
  g1[7] = 0;
  g2[0] = 1; g2[1] = 1; g2[2] = 0; g2[3] = 0;   // tensor_dim2=1, tensor_dim3=1
  g3[0] = 0; g3[1] = 0x10000; g3[2] = 0; g3[3] = 0; // tensor_dim4=1
}

__device__ __forceinline__ void tdm_load_1d(uint32_t lds_byte, const void* g, uint32_t ndw) {
  v4u g0; v8i g1; v4i g2, g3;
  tdm_desc(lds_byte, (uint64_t)(uintptr_t)g, ndw, g0, g1, g2, g3);
  v8i gx = {0, 0, 0, 0, 0, 0, 0, 0};            // extra descriptor group: unused
  __builtin_amdgcn_tensor_load_to_lds(g0, g1, g2, g3, gx, 0);
}

__device__ __forceinline__ void tdm_store_1d(uint32_t lds_byte, void* g, uint32_t ndw) {
  v4u g0; v8i g1; v4i g2, g3;
  tdm_desc(lds_byte, (uint64_t)(uintptr_t)g, ndw, g0, g1, g2, g3);
  v8i gx = {0, 0, 0, 0, 0, 0, 0, 0};
  __builtin_amdgcn_tensor_store_from_lds(g0, g1, g2, g3, gx, 0);
}

// ---------------- kernel ----------------

__global__ __launch_bounds__(32) void struc_tree_chain(
    const float* __restrict__ x,
    const float* __restrict__ Wmd, const float* __restrict__ bmd,
    const float* __restrict__ Wmu, const float* __restrict__ bmu,
    const float* __restrict__ Wud, const float* __restrict__ bud,
    const float* __restrict__ Wuu, const float* __restrict__ buu,
    float* __restrict__ out, float* __restrict__ hd_g) {
  __shared__ float lds[LDS_FLOATS];
  const int lane = threadIdx.x;
  const int col  = lane & 15;
  const int hi   = lane >> 4;        // 0 = lanes 0-15, 1 = lanes 16-31
  const bool lo  = (hi == 0);
  const bool canon = (col == 0);     // lanes 0 and 16 carry canonical D values
  const uint32_t lds_base = (uint32_t)(uintptr_t)(&lds[0]);

  // ---- preload weights (A-layout, W^T in 16x4 chunks) and biases (D-layout)
  v2f Amd[4], Amu[4], Aud[4], Auul[4], Auuh[4], Audx;
#pragma unroll
  for (int c = 0; c < 4; ++c) {
    int r0 = 4 * c + 2 * hi;
    Amd[c]  = (v2f){Wmd[(r0 + 0) * 16 + col],        Wmd[(r0 + 1) * 16 + col]};
    Amu[c]  = (v2f){Wmu[(r0 + 0) * 16 + col],        Wmu[(r0 + 1) * 16 + col]};
    Aud[c]  = (v2f){Wud[(2 + r0 + 0) * 16 + col],    Wud[(2 + r0 + 1) * 16 + col]};
    Auul[c] = (v2f){Wuu[(r0 + 0) * 16 + col],        Wuu[(r0 + 1) * 16 + col]};
    Auuh[c] = (v2f){Wuu[(16 + r0 + 0) * 16 + col],   Wuu[(16 + r0 + 1) * 16 + col]};
  }
  // padded K=16..19 chunk of update_down: rows {x0, x1, 0, 0}
  Audx = (v2f){hi ? 0.f : Wud[0 * 16 + col], hi ? 0.f : Wud[1 * 16 + col]};
  v8f bmdD, budD, bmuD, buuD;
#pragma unroll
  for (int r = 0; r < 8; ++r) {
    bmdD[r] = bmd[r + 8 * hi];
    budD[r] = bud[r + 8 * hi];
    bmuD[r] = bmu[r + 8 * hi];
    buuD[r] = buu[r + 8 * hi];
  }
  int belem[8];
#pragma unroll
  for (int v = 0; v < 8; ++v) belem[v] = 4 * (v >> 1) + (v & 1) + 2 * hi;

  // =================== DOWN PASS ===================
  float hB[8];
#pragma unroll
  for (int r = 0; r < 8; ++r) hB[r] = 0.f;

  tdm_load_1d(lds_base + X_OFF * 4, x, 2 * BSZ);  // prefetch x block 0

#pragma unroll 1
  for (int blk = 0; blk < NB; ++blk) {
    if (blk + 1 < NB)
      tdm_load_1d(lds_base + (X_OFF + ((blk + 1) & 1) * 512) * 4,
                  x + (size_t)(blk + 1) * BSZ * 2, 2 * BSZ);
    // In-order TENSORcnt: wait for everything older than the last K ops.
    if (blk == 0)            __builtin_amdgcn_s_wait_tensorcnt(1);
    else if (blk + 1 < NB)   __builtin_amdgcn_s_wait_tensorcnt(2);
    else                     __builtin_amdgcn_s_wait_tensorcnt(1);
    asm volatile("" ::: "memory");

    const int xbase  = X_OFF + (blk & 1) * 512;
    const int hdbase = HD_OFF + (blk & 1) * 4096;

#pragma unroll 1
    for (int j = 0; j < BSZ; ++j) {
      const int i = blk * BSZ + j;
      float x0 = lds[xbase + 2 * j];
      float x1 = lds[xbase + 2 * j + 1];
      // x-part + bias first (independent of the message chain)
      v2f Bx; Bx.x = hi ? 0.f : x0; Bx.y = hi ? 0.f : x1;
      v8f acc = __builtin_amdgcn_wmma_f32_16x16x4_f32(false, Audx, false, Bx,
                                                      (short)0, budD, false, false);
      if (i != 0) {
        v8f mD = wmma4(Amd, hB, bmdD);        // message_down pre-activation
        float msD[8], mB[8];
        sigB(mD, lo, msD, mB);                // sigmoid + relayout to B
        acc = wmma4(Aud, mB, acc);            // message part of update_down
      }
      float sD[8];
      sigB(acc, lo, sD, hB);                  // h_i; hB feeds next step
      const int nodebase = hdbase + j * 16 + 8 * hi;
#pragma unroll
      for (int r = 0; r < 8; ++r)             // conflict-free scatter to tile
        lds[canon ? (nodebase + r) : (DUMMY_OFF + lane)] = sD[r];
    }
    asm volatile("s_wait_dscnt 0x0" ::: "memory");
    tdm_store_1d(lds_base + (uint32_t)(HD_OFF + (blk & 1) * 4096) * 4,
                 hd_g + (size_t)blk * BSZ * 16, 16 * BSZ);
  }

  // =================== UP PASS ===================
  float gB[8];
#pragma unroll
  for (int r = 0; r < 8; ++r) gB[r] = 0.f;

  // prefetch last hd block (TDM is in-order: all down-pass stores complete first)
  tdm_load_1d(lds_base + (uint32_t)(HD_OFF + ((NB - 1) & 1) * 4096) * 4,
              hd_g + (size_t)(NB - 1) * BSZ * 16, 16 * BSZ);

#pragma unroll 1
  for (int blk = NB - 1; blk >= 0; --blk) {
    if (blk > 0)
      tdm_load_1d(lds_base + (uint32_t)(HD_OFF + ((blk - 1) & 1) * 4096) * 4,
                  hd_g + (size_t)(blk - 1) * BSZ * 16, 16 * BSZ);
    if (blk > 0) __builtin_amdgcn_s_wait_tensorcnt(1);
    else         __builtin_amdgcn_s_wait_tensorcnt(0);
    asm volatile("" ::: "memory");

    const int hdbase = HD_OFF + (blk & 1) * 4096;

#pragma unroll 1
    for (int j = BSZ - 1; j >= 0; --j) {
      const int i = blk * BSZ + j;
      float hdB[8];
#pragma unroll
      for (int v = 0; v < 8; ++v)             // broadcast ds_loads, B-layout
        hdB[v] = lds[hdbase + j * 16 + belem[v]];
      v8f acc = wmma4(Auul, hdB, buuD);       // hd part + bias first
      if (i != NNODE - 1) {
        v8f mD = wmma4(Amu, gB, bmuD);        // message_up pre-activation
        float msD[8], mB[8];
        sigB(mD, lo, msD, mB);
        acc = wmma4(Auuh, mB, acc);           // message part of update_up
      }
      float sD[8];
      sigB(acc, lo, sD, gB);
      if (i == 0) {                           // g_root -> d_out (16 floats)
#pragma unroll
        for (int r = 0; r < 8; ++r)
          if (canon) out[r + 8 * hi] = sD[r];
      }
    }
  }
}

// ---------------- host launch ----------------

extern "C" void kernel_launch(void* const* d_in, const int* in_sizes, int n_in,
                              void* d_out, int out_size, void* d_ws, size_t ws_size,
                              hipStream_t stream) {
  const float* x   = (const float*)d_in[0];
  const float* Wmd = (const float*)d_in[1];
  const float* bmd = (const float*)d_in[2];
  const float* Wmu = (const float*)d_in[3];
  const float* bmu = (const float*)d_in[4];
  const float* Wud = (const float*)d_in[5];
  const float* bud = (const float*)d_in[6];
  const float* Wuu = (const float*)d_in[7];
  const float* buu = (const float*)d_in[8];
  // d_in[9] = num_node (compile-time constant here), d_in[10] = edge_index (ignored by reference)
  float* hd = (float*)d_ws;  // N*16 f32 = 32 MB scratch for the down-pass states
  struc_tree_chain<<<1, 32, 0, stream>>>(x, Wmd, bmd, Wmu, bmu, Wud, bud,
                                         Wuu, buu, (float*)d_out, hd);
  (void)in_sizes; (void)n_in; (void)out_size; (void)ws_size;
}